// Q_NETWORK_42356967473573
// MI455X (gfx1250) — compile-verified
//
#include <hip/hip_runtime.h>
#include <cstddef>

#define N_NODES 50000
#define N_EDGES 200000
#define KNBR    4
#define DE      6
#define DEH     24
#define DH      28
#define DG      112      // 4*DH
#define LALPHA  0.3f

typedef __attribute__((ext_vector_type(2))) float v2f;
typedef __attribute__((ext_vector_type(8))) float v8f;

// ---------------------------------------------------------------------------
// f32 WMMA helper: D(16x16) = A(16x4) * B(4x16) + C
// A layout : lane L holds row M=L%16 ; VGPR0=K(khalf+0), VGPR1=K(khalf+1),
//            khalf = (L>=16)?2:0
// B layout : lane L holds col N=L%16 ; VGPR0=K(khalf+0), VGPR1=K(khalf+1)
// C layout : lane L, VGPR v -> row M = v + 8*(L>=16), col N = L%16
// ---------------------------------------------------------------------------
__device__ __forceinline__ v8f wmma4(v2f a, v2f b, v8f c) {
  return __builtin_amdgcn_wmma_f32_16x16x4_f32(
      /*neg_a=*/false, a, /*neg_b=*/false, b,
      /*c_mod=*/(short)0, c, /*reuse_a=*/false, /*reuse_b=*/false);
}

__device__ __forceinline__ float relu_f(float x) { return x > 0.0f ? x : 0.0f; }
__device__ __forceinline__ float sigmoid_f(float x) { return 1.0f / (1.0f + expf(-x)); }

// ---------------------------------------------------------------------------
// Kernel 0: zero the segment-sum accumulator (ws poisoned by harness)
// ---------------------------------------------------------------------------
__global__ __launch_bounds__(256) void zero_agg(float* __restrict__ agg, int n) {
  int i = blockIdx.x * 256 + threadIdx.x;
  if (i < n) agg[i] = 0.0f;
}

// ---------------------------------------------------------------------------
// Kernel 1: edge MLP (two relu layers via V_WMMA_F32_16X16X4_F32) + atomic
// scatter-add into agg. 8 waves / block, each wave owns 16 edges.
// Weights zero-padded into LDS once per block -> branch-free B fragments.
// ---------------------------------------------------------------------------
__global__ __launch_bounds__(256) void edge_mlp_scatter(
    const float* __restrict__ ef, const int* __restrict__ edge_node,
    const float* __restrict__ W1, const float* __restrict__ b1,
    const float* __restrict__ W2, const float* __restrict__ b2,
    float* __restrict__ agg)
{
  __shared__ float W1s[8][32];       // K padded 6->8, cols padded 24->32
  __shared__ float W2s[24][32];      // cols padded 24->32
  __shared__ float b1s[32];
  __shared__ float b2s[32];
  __shared__ float As [8][16][8];    // per-wave: 16 edges x K(6->8)
  __shared__ float Mid[8][16][32];   // per-wave layer-1 out (cols padded)
  __shared__ float Out[8][16][32];   // per-wave layer-2 out (cols padded)

  const int tid  = threadIdx.x;
  const int wv   = tid >> 5;
  const int lane = tid & 31;
  const int e0   = (blockIdx.x * 8 + wv) * 16;

  // ---- stage padded weights (block-wide) ----
  for (int idx = tid; idx < 8 * 32; idx += 256) {
    const int r = idx >> 5, c = idx & 31;
    W1s[r][c] = (r < DE && c < DEH) ? W1[r * DEH + c] : 0.0f;
  }
  for (int idx = tid; idx < 24 * 32; idx += 256) {
    const int r = idx >> 5, c = idx & 31;
    W2s[r][c] = (c < DEH) ? W2[r * DEH + c] : 0.0f;
  }
  if (tid < 32) {
    b1s[tid] = (tid < DEH) ? b1[tid] : 0.0f;
    b2s[tid] = (tid < DEH) ? b2[tid] : 0.0f;
  }

  // ---- stage this wave's 16x6 edge-feature tile (zero padded) ----
  for (int idx = lane; idx < 16 * 8; idx += 32) {
    const int r = idx >> 3, k = idx & 7;
    const int e = e0 + r;
    As[wv][r][k] = (k < DE && e < N_EDGES) ? ef[(size_t)e * DE + k] : 0.0f;
  }
  __syncthreads();

  const int m     = lane & 15;          // A row / B,C column within tile
  const int khalf = (lane >> 4) << 1;   // 0 or 2
  const int crow0 = (lane >> 4) << 3;   // C row base: 0 or 8

  // ---- layer 1: (16x6) @ (6x24) ----
#pragma unroll
  for (int t = 0; t < 2; ++t) {
    const int col = t * 16 + m;
    v8f c = {};
#pragma unroll
    for (int k0 = 0; k0 < 8; k0 += 4) {
      const int ka = k0 + khalf;
      v2f a, b;
      a[0] = As[wv][m][ka];  a[1] = As[wv][m][ka + 1];
      b[0] = W1s[ka][col];   b[1] = W1s[ka + 1][col];
      c = wmma4(a, b, c);
    }
    const float bias = b1s[col];
#pragma unroll
    for (int v = 0; v < 8; ++v)
      Mid[wv][crow0 + v][col] = relu_f(c[v] + bias);
  }
  __syncthreads();

  // ---- layer 2: (16x24) @ (24x24), K = 24 -> 6 k-steps ----
#pragma unroll
  for (int t = 0; t < 2; ++t) {
    const int col = t * 16 + m;
    v8f c = {};
#pragma unroll
    for (int k0 = 0; k0 < DEH; k0 += 4) {
      const int ka = k0 + khalf;
      v2f a, b;
      a[0] = Mid[wv][m][ka]; a[1] = Mid[wv][m][ka + 1];
      b[0] = W2s[ka][col];   b[1] = W2s[ka + 1][col];
      c = wmma4(a, b, c);
    }
    const float bias = b2s[col];
#pragma unroll
    for (int v = 0; v < 8; ++v)
      Out[wv][crow0 + v][col] = relu_f(c[v] + bias);
  }
  __syncthreads();

  // ---- segment-sum scatter ----
  for (int idx = lane; idx < 16 * DEH; idx += 32) {
    const int r = idx / DEH, col = idx % DEH;
    const int e = e0 + r;
    if (e < N_EDGES) {
      const int node = edge_node[e];
      atomicAdd(&agg[(size_t)node * DEH + col], Out[wv][r][col]);
    }
  }
}

// ---------------------------------------------------------------------------
// Kernel 2: node_vec = relu([agg | onehot(phase)] @ W_ne + b_ne)  (K=28,N=28)
// 8 waves / block, each wave owns 16 nodes; W_ne zero-padded in LDS.
// ---------------------------------------------------------------------------
__global__ __launch_bounds__(256) void node_enc(
    const float* __restrict__ agg, const int* __restrict__ phase,
    const float* __restrict__ Wne, const float* __restrict__ bne,
    float* __restrict__ node_vec)
{
  __shared__ float Ws[DH][32];        // 28 x (28 padded to 32)
  __shared__ float bs[32];
  __shared__ float X[8][16][DH];      // per-wave input tile

  const int tid  = threadIdx.x;
  const int wv   = tid >> 5;
  const int lane = tid & 31;
  const int r0   = (blockIdx.x * 8 + wv) * 16;

  for (int idx = tid; idx < DH * 32; idx += 256) {
    const int r = idx >> 5, c = idx & 31;
    Ws[r][c] = (c < DH) ? Wne[r * DH + c] : 0.0f;
  }
  if (tid < 32) bs[tid] = (tid < DH) ? bne[tid] : 0.0f;

  for (int idx = lane; idx < 16 * DH; idx += 32) {
    const int r = idx / DH, k = idx % DH;
    const int node = r0 + r;
    float v = 0.0f;
    if (node < N_NODES) {
      if (k < DEH) v = agg[(size_t)node * DEH + k];
      else         v = (phase[node] == (k - DEH)) ? 1.0f : 0.0f;
    }
    X[wv][r][k] = v;
  }
  __syncthreads();

  const int m     = lane & 15;
  const int khalf = (lane >> 4) << 1;
  const int crow0 = (lane >> 4) << 3;

#pragma unroll
  for (int t = 0; t < 2; ++t) {
    const int col = t * 16 + m;
    v8f c = {};
#pragma unroll
    for (int k0 = 0; k0 < DH; k0 += 4) {          // K = 28 -> 7 k-steps
      const int ka = k0 + khalf;
      v2f a, b;
      a[0] = X[wv][m][ka]; a[1] = X[wv][m][ka + 1];
      b[0] = Ws[ka][col];  b[1] = Ws[ka + 1][col];
      c = wmma4(a, b, c);
    }
    if (col < DH) {
      const float bias = bs[col];
#pragma unroll
      for (int v = 0; v < 8; ++v) {
        const int node = r0 + crow0 + v;
        if (node < N_NODES)
          node_vec[(size_t)node * DH + col] = relu_f(c[v] + bias);
      }
    }
  }
}

// ---------------------------------------------------------------------------
// Kernel 3: per-node LSTM (HBM-bound: streams Wx/Wh once, non-temporal).
// One block of 128 threads per node; threads 0..111 each own one gate column.
// ---------------------------------------------------------------------------
__global__ __launch_bounds__(128) void lstm_stream(
    const float* __restrict__ node_vec, const float* __restrict__ h0,
    const float* __restrict__ c0,
    const float* __restrict__ Wx, const float* __restrict__ Wh,
    const float* __restrict__ b_lstm,
    float* __restrict__ h_out, float* __restrict__ c_out)
{
  __shared__ float nv[DH];
  __shared__ float hp[DH];
  __shared__ float z[DG];

  const int n = blockIdx.x;
  const int t = threadIdx.x;

  if (t < DH)            nv[t]      = node_vec[(size_t)n * DH + t];
  else if (t < 2 * DH)   hp[t - DH] = h0[(size_t)n * DH + (t - DH)];
  __syncthreads();

  if (t < DG) {
    const float* wx = Wx + (size_t)n * DH * DG + t;
    const float* wh = Wh + (size_t)n * DH * DG + t;
    float acc = b_lstm[(size_t)n * DG + t];
#pragma unroll 7
    for (int d = 0; d < DH; ++d) {
      acc += nv[d] * __builtin_nontemporal_load(wx + (size_t)d * DG);
      acc += hp[d] * __builtin_nontemporal_load(wh + (size_t)d * DG);
    }
    z[t] = acc;
  }
  __syncthreads();

  if (t < DH) {
    const float zi = z[t], zf = z[DH + t], zg = z[2 * DH + t], zo = z[3 * DH + t];
    const float c = sigmoid_f(zf) * c0[(size_t)n * DH + t] +
                    sigmoid_f(zi) * tanhf(zg);
    const float h = sigmoid_f(zo) * tanhf(c);
    c_out[(size_t)n * DH + t] = c;
    h_out[(size_t)n * DH + t] = h;
  }
}

// ---------------------------------------------------------------------------
// Kernel 4: neighbor attention + Q head. One wave32 per node, 4 nodes/block.
// ---------------------------------------------------------------------------
__device__ __forceinline__ float wave_sum32(float v) {
#pragma unroll
  for (int off = 16; off >= 1; off >>= 1) v += __shfl_xor(v, off, 32);
  return v;
}

__global__ __launch_bounds__(128) void att_qhead(
    const float* __restrict__ h_out, const int* __restrict__ nbr,
    const float* __restrict__ w_att,
    const float* __restrict__ Wq1, const float* __restrict__ bq1,
    const float* __restrict__ Wq2, const float* __restrict__ bq2,
    float* __restrict__ q_out)
{
  __shared__ float ov[4][2 * DH];   // [h | att] per wave
  __shared__ float hq[4][15];

  const int lane = threadIdx.x & 31;
  const int w    = threadIdx.x >> 5;
  const int n    = blockIdx.x * 4 + w;              // N % 4 == 0
  const int d    = lane;
  const bool vd  = d < DH;

  const float hval = vd ? h_out[(size_t)n * DH + d] : 0.0f;
  const float wai  = vd ? w_att[d]       : 0.0f;
  const float waj  = vd ? w_att[DH + d]  : 0.0f;

  const float s_i = wave_sum32(hval * wai);

  float acc = 0.0f, denom = 0.0f;
#pragma unroll
  for (int k = 0; k < KNBR; ++k) {
    const int j   = nbr[(size_t)n * KNBR + k];
    const float hj = vd ? h_out[(size_t)j * DH + d] : 0.0f;
    float s = s_i + wave_sum32(hj * waj);
    s = (s > 0.0f) ? s : LALPHA * s;                // leaky_relu(0.3)
    const float ex = expf(s);
    acc   += ex * hj;
    denom += ex;
  }

  if (vd) {
    ov[w][d]      = hval;
    ov[w][DH + d] = acc / denom;
  }
  __syncthreads();

  if (lane < 15) {
    float a = bq1[lane];
#pragma unroll 8
    for (int m = 0; m < 2 * DH; ++m) a += ov[w][m] * Wq1[m * 15 + lane];
    hq[w][lane] = relu_f(a);
  }
  __syncthreads();

  if (lane < 2) {
    float a = bq2[lane];
#pragma unroll
    for (int m = 0; m < 15; ++m) a += hq[w][m] * Wq2[m * 2 + lane];
    q_out[(size_t)n * 2 + lane] = relu_f(a);
  }
}

// ---------------------------------------------------------------------------
// Host launcher
// ---------------------------------------------------------------------------
extern "C" void kernel_launch(void* const* d_in, const int* in_sizes, int n_in,
                              void* d_out, int out_size, void* d_ws, size_t ws_size,
                              hipStream_t stream) {
  const float* edge_feats = (const float*)d_in[0];
  const int*   phase      = (const int*)  d_in[1];
  const int*   edge_node  = (const int*)  d_in[2];
  const int*   nbr_idx    = (const int*)  d_in[3];
  const float* h0         = (const float*)d_in[4];
  const float* c0         = (const float*)d_in[5];
  const float* W_e1       = (const float*)d_in[6];
  const float* b_e1       = (const float*)d_in[7];
  const float* W_e2       = (const float*)d_in[8];
  const float* b_e2       = (const float*)d_in[9];
  const float* W_ne       = (const float*)d_in[10];
  const float* b_ne       = (const float*)d_in[11];
  const float* Wx         = (const float*)d_in[12];
  const float* Wh         = (const float*)d_in[13];
  const float* b_lstm     = (const float*)d_in[14];
  const float* w_att      = (const float*)d_in[15];
  const float* W_q1       = (const float*)d_in[16];
  const float* b_q1       = (const float*)d_in[17];
  const float* W_q2       = (const float*)d_in[18];
  const float* b_q2       = (const float*)d_in[19];
  (void)in_sizes; (void)n_in; (void)out_size; (void)ws_size;

  float* out   = (float*)d_out;
  float* q_out = out;                          // N*2
  float* h_out = out + (size_t)2  * N_NODES;   // N*28
  float* c_out = out + (size_t)30 * N_NODES;   // N*28

  float* agg      = (float*)d_ws;                   // N*24 floats
  float* node_vec = agg + (size_t)N_NODES * DEH;    // N*28 floats

  const int aggN = N_NODES * DEH;
  zero_agg<<<(aggN + 255) / 256, 256, 0, stream>>>(agg, aggN);

  const int edgeWaves = N_EDGES / 16;               // 12500
  edge_mlp_scatter<<<(edgeWaves + 7) / 8, 256, 0, stream>>>(
      edge_feats, edge_node, W_e1, b_e1, W_e2, b_e2, agg);

  const int nodeWaves = (N_NODES + 15) / 16;        // 3125
  node_enc<<<(nodeWaves + 7) / 8, 256, 0, stream>>>(
      agg, phase, W_ne, b_ne, node_vec);

  lstm_stream<<<N_NODES, 128, 0, stream>>>(
      node_vec, h0, c0, Wx, Wh, b_lstm, h_out, c_out);

  att_qhead<<<N_NODES / 4, 128, 0, stream>>>(
      h_out, nbr_idx, w_att, W_q1, b_q1, W_q2, b_q2, q_out);
}